// OutCount_23579370455451
// MI455X (gfx1250) — compile-verified
//
#include <hip/hip_runtime.h>
#include <math.h>

// Problem constants (match the reference).
#define L_SEQ 512
#define T_SEQ 64
#define B_N   2048
#define D_DIM 512
#define K_DIM 1024   // 2*D
#define BN_EPS 1e-5f

typedef float v2f __attribute__((ext_vector_type(2)));
typedef float v4f __attribute__((ext_vector_type(4)));
typedef float v8f __attribute__((ext_vector_type(8)));

__device__ __forceinline__ float gelu_erf(float x) {
    // torch nn.GELU default = exact erf formulation
    return 0.5f * x * (1.0f + erff(x * 0.70710678118654752f));
}

// ---------------------------------------------------------------------------
// Kernel 1: ragged masked mean over L (per-b length) + mean over T.
// Grid = B blocks, 128 threads, each thread owns a float4 slice of D.
// Fully coalesced 2KB row reads; streams the 2.4 GB of input at HBM rate.
// ---------------------------------------------------------------------------
__global__ __launch_bounds__(128) void k_means(
    const float* __restrict__ qv, const float* __restrict__ vq,
    const int* __restrict__ qlen, float* __restrict__ X)
{
    const int b  = blockIdx.x;
    const int t4 = threadIdx.x * 4;        // 0..508
    const int len = qlen[b];               // >= 1 per reference

    // --- question mean: sum only the unmasked rows (l < len) ---
    const float* base = qv + (size_t)b * D_DIM + t4;
    v4f acc = {0.f, 0.f, 0.f, 0.f};
    for (int l = 0; l < len; ++l) {
        const float* p = base + (size_t)l * (size_t)B_N * D_DIM;
        if (l + 1 < len)
            __builtin_prefetch(p + (size_t)B_N * D_DIM, 0, 3); // global_prefetch_b8
        acc += *(const v4f*)p;
    }
    acc *= (1.0f / (float)len);
    *(v4f*)(X + (size_t)b * K_DIM + t4) = acc;

    // --- visual mean over T=64 ---
    const float* vb = vq + (size_t)b * D_DIM + t4;
    v4f acc2 = {0.f, 0.f, 0.f, 0.f};
    #pragma unroll 4
    for (int t = 0; t < T_SEQ; ++t) {
        acc2 += *(const v4f*)(vb + (size_t)t * (size_t)B_N * D_DIM);
    }
    acc2 *= (1.0f / (float)T_SEQ);
    *(v4f*)(X + (size_t)b * K_DIM + D_DIM + t4) = acc2;
}

// ---------------------------------------------------------------------------
// Kernel 2: h = gelu(x @ W1^T + b1) via V_WMMA_F32_16X16X4_F32 (fp32-exact).
// One wave per 16x16 output tile, K=1024 in steps of 4.
// A (16x4 f32):  lane = 16*(k_half) + m, vgpr v -> k = 2*(lane/16)+v
// B (4x16 f32):  lane = 16*(k_half) + n, vgpr v -> k = 2*(lane/16)+v
// C/D (16x16):   vgpr r, lane -> m = r + 8*(lane/16), n = lane%16
// ---------------------------------------------------------------------------
__global__ __launch_bounds__(128) void k_gemm_gelu(
    const float* __restrict__ X, const float* __restrict__ W1,
    const float* __restrict__ b1, float* __restrict__ H)
{
    const int lane = threadIdx.x & 31;
    const int wave = threadIdx.x >> 5;
    const int half = lane >> 4;
    const int l16  = lane & 15;

    const int tile_n = blockIdx.x;               // 0..31  (j tiles)
    const int tile_m = blockIdx.y * 4 + wave;    // 0..127 (b tiles)

    const int j   = tile_n * 16 + l16;           // output feature (B-matrix n)
    const int row = tile_m * 16 + l16;           // batch row      (A-matrix m)

    const float* aptr = X  + (size_t)row * K_DIM + 2 * half;  // 8B-aligned
    const float* bptr = W1 + (size_t)j   * K_DIM + 2 * half;  // W1[j, k]

    v8f c = {0.f, 0.f, 0.f, 0.f, 0.f, 0.f, 0.f, 0.f};
    #pragma unroll 8
    for (int k = 0; k < K_DIM; k += 4) {
        v2f a  = *(const v2f*)(aptr + k);
        v2f bb = *(const v2f*)(bptr + k);
        // 8 args: (neg_a, A, neg_b, B, c_mod, C, reuse_a, reuse_b)
        c = __builtin_amdgcn_wmma_f32_16x16x4_f32(false, a, false, bb,
                                                  (short)0, c, false, false);
    }

    const float bias  = b1[j];
    const int   mbase = tile_m * 16 + 8 * half;
    #pragma unroll
    for (int r = 0; r < 8; ++r) {
        H[(size_t)(mbase + r) * D_DIM + j] = gelu_erf(c[r] + bias);
    }
}

// ---------------------------------------------------------------------------
// Kernel 3: BatchNorm batch statistics per feature (biased variance).
// Grid = D blocks; block reduces column j over B=2048 rows. h is L2-resident.
// ---------------------------------------------------------------------------
__global__ __launch_bounds__(256) void k_colstats(
    const float* __restrict__ H, float* __restrict__ MU, float* __restrict__ RSTD)
{
    const int j = blockIdx.x;
    const int t = threadIdx.x;
    float s = 0.f, s2 = 0.f;
    for (int r = t; r < B_N; r += 256) {
        float v = H[(size_t)r * D_DIM + j];
        s += v; s2 += v * v;
    }
    __shared__ float sh[256], sh2[256];
    sh[t] = s; sh2[t] = s2;
    __syncthreads();
    for (int off = 128; off > 0; off >>= 1) {
        if (t < off) { sh[t] += sh[t + off]; sh2[t] += sh2[t + off]; }
        __syncthreads();
    }
    if (t == 0) {
        float mu  = sh[0]  * (1.0f / (float)B_N);
        float var = sh2[0] * (1.0f / (float)B_N) - mu * mu;  // biased (ddof=0)
        MU[j]   = mu;
        RSTD[j] = rsqrtf(var + BN_EPS);
    }
}

// ---------------------------------------------------------------------------
// Kernel 4: fold BN affine + W2 into per-feature scale s[j] and scalar C:
//   out[b] = sum_j s[j]*h[b,j] + C
//   s[j] = W2[j]*gamma[j]*rstd[j]
//   C    = b2 + sum_j W2[j]*(beta[j] - mu[j]*gamma[j]*rstd[j])
// ---------------------------------------------------------------------------
__global__ __launch_bounds__(512) void k_coeffs(
    const float* __restrict__ MU, const float* __restrict__ RSTD,
    const float* __restrict__ gamma, const float* __restrict__ beta,
    const float* __restrict__ W2, const float* __restrict__ b2,
    float* __restrict__ S, float* __restrict__ CC)
{
    const int j = threadIdx.x;                  // 0..511
    float g  = gamma[j] * RSTD[j];
    float w  = W2[j];
    S[j] = w * g;
    __shared__ float sh[512];
    sh[j] = w * (beta[j] - MU[j] * g);
    __syncthreads();
    for (int off = 256; off > 0; off >>= 1) {
        if (j < off) sh[j] += sh[j + off];
        __syncthreads();
    }
    if (j == 0) CC[0] = sh[0] + b2[0];
}

// ---------------------------------------------------------------------------
// Kernel 5: out[b] = dot(s, h[b,:]) + C. One wave32 per row, shfl reduction.
// ---------------------------------------------------------------------------
__global__ __launch_bounds__(256) void k_out(
    const float* __restrict__ H, const float* __restrict__ S,
    const float* __restrict__ CC, float* __restrict__ out)
{
    const int lane = threadIdx.x & 31;
    const int wave = threadIdx.x >> 5;
    const int b = blockIdx.x * 8 + wave;
    const float* hrow = H + (size_t)b * D_DIM;
    float acc = 0.f;
    #pragma unroll
    for (int i = 0; i < 16; ++i) {
        const int idx = lane + 32 * i;
        acc += hrow[idx] * S[idx];
    }
    #pragma unroll
    for (int off = 16; off > 0; off >>= 1)
        acc += __shfl_xor(acc, off, 32);
    if (lane == 0) out[b] = acc + CC[0];
}

// ---------------------------------------------------------------------------
extern "C" void kernel_launch(void* const* d_in, const int* in_sizes, int n_in,
                              void* d_out, int out_size, void* d_ws, size_t ws_size,
                              hipStream_t stream)
{
    const float* qv    = (const float*)d_in[0];  // [L,B,D]
    const float* vq    = (const float*)d_in[1];  // [T,B,D]
    const int*   qlen  = (const int*)  d_in[2];  // [B]
    const float* W1    = (const float*)d_in[3];  // [D, 2D]
    const float* b1    = (const float*)d_in[4];  // [D]
    const float* gamma = (const float*)d_in[5];  // [D]
    const float* beta  = (const float*)d_in[6];  // [D]
    const float* W2    = (const float*)d_in[7];  // [1, D]
    const float* b2    = (const float*)d_in[8];  // [1]
    float* out = (float*)d_out;                  // [B, 1]

    // Workspace layout (floats): X[B*2D] | H[B*D] | MU[D] | RSTD[D] | S[D] | C[1]
    float* ws   = (float*)d_ws;
    float* X    = ws;
    float* H    = X + (size_t)B_N * K_DIM;
    float* MU   = H + (size_t)B_N * D_DIM;
    float* RSTD = MU + D_DIM;
    float* S    = RSTD + D_DIM;
    float* CC   = S + D_DIM;

    k_means    <<<B_N, 128, 0, stream>>>(qv, vq, qlen, X);
    k_gemm_gelu<<<dim3(32, 32), 128, 0, stream>>>(X, W1, b1, H);
    k_colstats <<<D_DIM, 256, 0, stream>>>(H, MU, RSTD);
    k_coeffs   <<<1, 512, 0, stream>>>(MU, RSTD, gamma, beta, W2, b2, S, CC);
    k_out      <<<B_N / 8, 256, 0, stream>>>(H, S, CC, out);
}